// HINGCN_18923625906522
// MI455X (gfx1250) — compile-verified
//
#include <hip/hip_runtime.h>
#include <hip/hip_bf16.h>

// Problem sizes (fixed by the reference)
#define NNODE   8192
#define NFEAT   128
#define NHID    32
#define DIMMP   32
#define NMETA   3
#define NCLS    16
#define LRELU_A 0.2f

typedef float v2f __attribute__((ext_vector_type(2)));
typedef float v8f __attribute__((ext_vector_type(8)));

#define KCHUNK          256   // K-values of B staged in LDS per chunk (32 KB)
#define WAVES_PER_BLOCK 8     // 256 threads / wave32
#define ROWS_PER_WAVE   32    // 2 x 16-row WMMA tiles per wave

// ---------------------------------------------------------------------------
// Generic tiled GEMM:  C[m] = act(A[m] (MxK) @ B[m] (Kx32) + bias[m])
//   - one wave computes a 32x32 output tile: 4 x V_WMMA_F32_16X16X4_F32 / step
//   - B panel chunk-staged in LDS via GLOBAL_LOAD_ASYNC_TO_LDS_B128,
//     shared by all 8 waves of the block (all waves work the same metapath)
//   - A streamed from global with per-lane b64 loads + prefetch; K-loop
//     unrolled x2 so multiple A loads are in flight per wait
// M is always NNODE (8192), ldb/ldc are always 32.
// ---------------------------------------------------------------------------
__launch_bounds__(256)
__global__ void gemm_wmma_f32(const float* __restrict__ Abase, long long strideAm,
                              const float* __restrict__ Bbase, long long strideBm,
                              float* __restrict__ Cbase, long long strideCm,
                              int lda, int K,
                              const float* __restrict__ biasBase, int strideBiasm,
                              int do_relu)
{
    __shared__ float Bs[KCHUNK * 32];   // 32 KB

    const int m    = blockIdx.y;
    const int tid  = threadIdx.x;
    const int wave = tid >> 5;
    const int lane = tid & 31;
    const int r0   = (blockIdx.x * WAVES_PER_BLOCK + wave) * ROWS_PER_WAVE;

    const float* __restrict__ A = Abase + (long long)m * strideAm;
    const float* __restrict__ B = Bbase + (long long)m * strideBm;
    float*       __restrict__ C = Cbase + (long long)m * strideCm;

    // A-fragment addressing (16x4 f32 layout):
    //   lane L: M = r0 + (L&15), K pair offset = 2*(L>>4)  -> one b64 load / step
    const int mrow = r0 + (lane & 15);
    const int koff = (lane >> 4) << 1;        // 0 or 2
    const int ncol = lane & 15;               // N within a 16-wide tile
    const float2* __restrict__ arow0 = (const float2*)(A + (long long)mrow * lda);
    const float2* __restrict__ arow1 = (const float2*)(A + (long long)(mrow + 16) * lda);

    v8f acc00 = {};   // rows [r0..r0+15],  N-tile [0..15]
    v8f acc01 = {};   // rows [r0..r0+15],  N-tile [16..31]
    v8f acc10 = {};   // rows [r0+16..+31], N-tile [0..15]
    v8f acc11 = {};   // rows [r0+16..+31], N-tile [16..31]

    const unsigned ldsBase = (unsigned)(uintptr_t)(void*)&Bs[0];

    for (int kc = 0; kc < K; kc += KCHUNK) {
        const int kcur = (K - kc) < KCHUNK ? (K - kc) : KCHUNK;

        // Async stage of B[kc : kc+kcur, 0:32] directly into LDS (no VGPR hop)
        {
            const float4* __restrict__ bsrc = (const float4*)(B + (long long)kc * 32);
            const int n4 = kcur * 8;                      // float4s in chunk
            for (int i = tid; i < n4; i += 256) {
                unsigned loff = ldsBase + (unsigned)i * 16u;
                const float4* gptr = bsrc + i;
                asm volatile("global_load_async_to_lds_b128 %0, %1, off"
                             :: "v"(loff), "v"(gptr) : "memory");
            }
            asm volatile("s_wait_asynccnt 0x0" ::: "memory");
        }
        __syncthreads();

        #pragma unroll 2
        for (int k0 = 0; k0 < kcur; k0 += 4) {
            // A fragments: {A[row][k+koff], A[row][k+koff+1]}  (8B aligned)
            const int kf = (kc + k0 + koff) >> 1;
            float2 a2l = arow0[kf];
            float2 a2h = arow1[kf];
            v2f af0; af0.x = a2l.x; af0.y = a2l.y;
            v2f af1; af1.x = a2h.x; af1.y = a2h.y;

            // stream-ahead prefetch of the A rows (512 B ahead)
            __builtin_prefetch((const void*)(arow0 + kf + 64), 0, 1);
            __builtin_prefetch((const void*)(arow1 + kf + 64), 0, 1);

            // B fragments from LDS: VGPR0 = row (k+koff), VGPR1 = row (k+koff+1)
            const float* bk = &Bs[(k0 + koff) * 32 + ncol];
            v2f b0; b0.x = bk[0];  b0.y = bk[32];        // N-tile 0
            v2f b1; b1.x = bk[16]; b1.y = bk[48];        // N-tile 1

            acc00 = __builtin_amdgcn_wmma_f32_16x16x4_f32(false, af0, false, b0,
                                                          (short)0, acc00, false, false);
            acc01 = __builtin_amdgcn_wmma_f32_16x16x4_f32(false, af0, false, b1,
                                                          (short)0, acc01, false, false);
            acc10 = __builtin_amdgcn_wmma_f32_16x16x4_f32(false, af1, false, b0,
                                                          (short)0, acc10, false, false);
            acc11 = __builtin_amdgcn_wmma_f32_16x16x4_f32(false, af1, false, b1,
                                                          (short)0, acc11, false, false);
        }
        __syncthreads();
    }

    // Epilogue: C/D layout -> VGPR r holds M = r (lanes 0-15) / M = r+8 (16-31)
    float bv0 = 0.f, bv1 = 0.f;
    if (biasBase) {
        const float* bias = biasBase + (long long)m * strideBiasm;
        bv0 = bias[ncol];
        bv1 = bias[16 + ncol];
    }
    const int rbase = r0 + ((lane >> 4) << 3);
    #pragma unroll
    for (int r = 0; r < 8; ++r) {
        float v00 = acc00[r] + bv0;
        float v01 = acc01[r] + bv1;
        float v10 = acc10[r] + bv0;
        float v11 = acc11[r] + bv1;
        if (do_relu) {
            v00 = fmaxf(v00, 0.f); v01 = fmaxf(v01, 0.f);
            v10 = fmaxf(v10, 0.f); v11 = fmaxf(v11, 0.f);
        }
        float* crow0 = C + (long long)(rbase + r) * 32;
        float* crow1 = C + (long long)(rbase + r + 16) * 32;
        crow0[ncol]      = v00;
        crow0[16 + ncol] = v01;
        crow1[ncol]      = v10;
        crow1[16 + ncol] = v11;
    }
}

// ---------------------------------------------------------------------------
// Per-node epilogue: metapath attention softmax, combine, linear, log_softmax
// ---------------------------------------------------------------------------
__launch_bounds__(256)
__global__ void attn_final(const float* __restrict__ H2,
                           const float* __restrict__ avec,
                           const float* __restrict__ Wlin,
                           const float* __restrict__ blin,
                           float* __restrict__ out)
{
    __shared__ float sA[DIMMP];
    __shared__ float sW[DIMMP * NCLS];
    __shared__ float sB[NCLS];

    const int tid = threadIdx.x;
    for (int i = tid; i < DIMMP * NCLS; i += 256) sW[i] = Wlin[i];
    if (tid < DIMMP) sA[tid] = avec[tid];
    if (tid < NCLS)  sB[tid] = blin[tid];
    __syncthreads();

    const int n = blockIdx.x * 256 + tid;

    // load h2[m][n][:] (vectorized b128)
    float h[NMETA][DIMMP];
    #pragma unroll
    for (int m = 0; m < NMETA; ++m) {
        const float4* src = (const float4*)(H2 + ((long long)m * NNODE + n) * DIMMP);
        float4* dst = (float4*)h[m];
        #pragma unroll
        for (int i = 0; i < DIMMP / 4; ++i) dst[i] = src[i];
    }

    // e[m] = leaky_relu(h[m] . a); softmax over metapaths
    float e[NMETA];
    float mx = -3.0e38f;
    #pragma unroll
    for (int m = 0; m < NMETA; ++m) {
        float s = 0.f;
        #pragma unroll
        for (int d = 0; d < DIMMP; ++d) s += h[m][d] * sA[d];
        e[m] = (s > 0.f) ? s : LRELU_A * s;
        mx = fmaxf(mx, e[m]);
    }
    float sum = 0.f;
    #pragma unroll
    for (int m = 0; m < NMETA; ++m) { e[m] = __expf(e[m] - mx); sum += e[m]; }
    const float inv = 1.f / sum;

    float o[DIMMP];
    #pragma unroll
    for (int d = 0; d < DIMMP; ++d) {
        float s = 0.f;
        #pragma unroll
        for (int m = 0; m < NMETA; ++m) s += e[m] * h[m][d];
        o[d] = s * inv;
    }

    // logits = relu(o @ Wlin + blin); then log_softmax over classes
    float lg[NCLS];
    float mx2 = -3.0e38f;
    #pragma unroll
    for (int j = 0; j < NCLS; ++j) {
        float s = sB[j];
        #pragma unroll
        for (int d = 0; d < DIMMP; ++d) s += o[d] * sW[d * NCLS + j];
        s = fmaxf(s, 0.f);
        lg[j] = s;
        mx2 = fmaxf(mx2, s);
    }
    float se = 0.f;
    #pragma unroll
    for (int j = 0; j < NCLS; ++j) se += __expf(lg[j] - mx2);
    const float lse = mx2 + __logf(se);

    float* orow = out + (long long)n * NCLS;
    #pragma unroll
    for (int j = 0; j < NCLS; ++j) orow[j] = lg[j] - lse;
}

// ---------------------------------------------------------------------------
extern "C" void kernel_launch(void* const* d_in, const int* in_sizes, int n_in,
                              void* d_out, int out_size, void* d_ws, size_t ws_size,
                              hipStream_t stream)
{
    (void)in_sizes; (void)n_in; (void)out_size; (void)ws_size;

    const float* x    = (const float*)d_in[0];   // [N, NFEAT]
    const float* adjs = (const float*)d_in[1];   // [M, N, N]
    const float* W1   = (const float*)d_in[2];   // [M, NFEAT, NHID]
    const float* b1   = (const float*)d_in[3];   // [M, NHID]
    const float* W2   = (const float*)d_in[4];   // [M, NHID, DIMMP]
    const float* b2   = (const float*)d_in[5];   // [M, DIMMP]
    const float* av   = (const float*)d_in[6];   // [DIMMP]
    const float* Wl   = (const float*)d_in[7];   // [DIMMP, NCLS]
    const float* bl   = (const float*)d_in[8];   // [NCLS]
    float* out = (float*)d_out;                  // [N, NCLS]

    float* ws = (float*)d_ws;
    const long long perM = (long long)NNODE * NHID;          // 262144 floats
    float* P1 = ws;                                          // [M,N,NHID]
    float* H1 = ws + 1 * NMETA * perM;                       // [M,N,NHID]
    float* P2 = ws + 2 * NMETA * perM;                       // [M,N,DIMMP]
    float* H2 = ws + 3 * NMETA * perM;                       // [M,N,DIMMP]

    dim3 blk(256);
    dim3 grd(NNODE / ROWS_PER_WAVE / WAVES_PER_BLOCK, NMETA); // (32, 3)

    // P1[m] = X @ W1[m]
    gemm_wmma_f32<<<grd, blk, 0, stream>>>(x, 0LL,
                                           W1, (long long)NFEAT * NHID,
                                           P1, perM,
                                           NFEAT, NFEAT, nullptr, 0, 0);
    // H1[m] = relu(A[m] @ P1[m] + b1[m])     -- HBM-dominant pass 1
    gemm_wmma_f32<<<grd, blk, 0, stream>>>(adjs, (long long)NNODE * NNODE,
                                           P1, perM,
                                           H1, perM,
                                           NNODE, NNODE, b1, NHID, 1);
    // P2[m] = H1[m] @ W2[m]
    gemm_wmma_f32<<<grd, blk, 0, stream>>>(H1, perM,
                                           W2, (long long)NHID * DIMMP,
                                           P2, perM,
                                           NHID, NHID, nullptr, 0, 0);
    // H2[m] = relu(A[m] @ P2[m] + b2[m])     -- HBM-dominant pass 2
    gemm_wmma_f32<<<grd, blk, 0, stream>>>(adjs, (long long)NNODE * NNODE,
                                           P2, perM,
                                           H2, perM,
                                           NNODE, NNODE, b2, DIMMP, 1);
    // attention + combine + linear + log_softmax
    attn_final<<<dim3(NNODE / 256), blk, 0, stream>>>(H2, av, Wl, bl, out);
}